// CLSTM_cell_11845519803052
// MI455X (gfx1250) — compile-verified
//
#include <hip/hip_runtime.h>
#include <math.h>

// ---------------------------------------------------------------------------
// ConvLSTM cell for MI455X (gfx1250), fp32 WMMA path.
//   SEQ=32 steps, B=32, F=64, L=256, OC=128 -> gates C=512, K=5, pad=2
//   Per step: GEMM (M=512, N=8192, K=5*192=960 dense) via
//   v_wmma_f32_16x16x4_f32, then GroupNorm(16 groups), then LSTM pointwise.
//   - K ordered (tap, ic): every 16x16x4 step carries 4 real K values.
//   - B (im2col) staged transposed in LDS: each fragment = one ds_load_b64.
//   - 2 N-tiles per wave share every A fragment (halves L2 A-traffic/FLOP).
// ---------------------------------------------------------------------------

typedef float v2f __attribute__((ext_vector_type(2)));
typedef float v8f __attribute__((ext_vector_type(8)));

#define SEQ   32
#define BSZ   32
#define FIN   64
#define LEN   256
#define OC    128
#define C4    512            // 4*OC gate channels
#define CIN   192            // F + OC
#define KW    5
#define KTOT  (KW * CIN)     // 960
#define NGRP  16
#define CHG   32             // channels per group
#define GSLAB (CHG * LEN)    // 8192 contiguous floats per (b, group)

#define SROW  20             // LDS row stride (floats): conflict-free b64 reads

// workspace layout (floats)
#define OFF_PACKW 0
#define SZ_PACKW  (C4 * KTOT)              // 491520 (1.9 MB)
#define OFF_H     (OFF_PACKW + SZ_PACKW)
#define SZ_H      (BSZ * OC * LEN)         // 1048576 (4 MB)
#define OFF_C     (OFF_H + SZ_H)
#define OFF_GATES (OFF_C + SZ_H)
#define SZ_GATES  (BSZ * C4 * LEN)         // 4194304 (16 MB)
#define OFF_STATS (OFF_GATES + SZ_GATES)   // 512 * {mean, rstd}

// -------------------------------------------------------------------- pack W
// conv_w (512,192,5) -> packW[oc][kk*192 + ic]  (K reordered tap-major).
__global__ __launch_bounds__(256) void pack_weights(const float* __restrict__ w,
                                                    float* __restrict__ packW) {
  int idx = blockIdx.x * 256 + threadIdx.x;
  if (idx >= C4 * KTOT) return;
  int oc = idx / KTOT, rem = idx % KTOT;
  int kk = rem / CIN, ic = rem % CIN;
  packW[idx] = w[(oc * CIN + ic) * KW + kk];
}

__global__ __launch_bounds__(256) void zero_buf(float* __restrict__ p, int n) {
  int idx = blockIdx.x * 256 + threadIdx.x;
  if (idx < n) p[idx] = 0.0f;
}

// ------------------------------------------------------------- GEMM (conv)
// Block: 256 threads = 8 waves. Block tile: M=128 (one 16-row A tile / wave),
// N=32 (two 16-col C tiles per wave; im2col staged once in LDS, transposed).
// grid = (512/128, 8192/32) = (4, 256).
// K loop: 12 chunks of 16 input channels; per chunk 5 taps x 4 ic-quads
// x 2 N-tiles = 40 wmma, all with fully dense K.
__global__ __launch_bounds__(256) void conv_gates_wmma(
    const float* __restrict__ x,      // inputs (SEQ,B,F,L)
    const float* __restrict__ h,      // h state (B,OC,L)
    const float* __restrict__ packW,  // (512, 960) tap-major K
    const float* __restrict__ bias,   // (512)
    float* __restrict__ gates,        // (B, 512, L)
    int t) {
  // transposed im2col tile: sBT[col][icRow], col 0..35 <-> l0-2 .. l0+33
  __shared__ float sBT[36][SROW];

  const int tid  = threadIdx.x;
  const int lane = tid & 31;
  const int wv   = tid >> 5;          // wave id 0..7
  const int hi   = lane >> 4;         // 0: K lo half, 1: K hi half
  const int n    = lane & 15;         // column within tile / A row
  const int m0   = blockIdx.x * 128 + wv * 16;
  const int nt   = blockIdx.y;
  const int b    = nt >> 3;
  const int l0   = (nt & 7) << 5;

  const float* xt = x + ((size_t)t * BSZ + b) * FIN * LEN;  // x[t][b][.][.]
  const float* hb = h + (size_t)b * OC * LEN;

  const float* wrow = packW + (size_t)(m0 + n) * KTOT;      // A row for lane
  const float* lds  = &sBT[0][0];

  v8f acc0 = {};
  v8f acc1 = {};

  for (int icc = 0; icc < CIN; icc += 16) {
    // each 16-channel chunk is homogeneous (F=64 is a multiple of 16):
    const float* srcs =
        (icc < FIN) ? (xt + (size_t)icc * LEN) : (hb + (size_t)(icc - FIN) * LEN);
    __syncthreads();
#pragma unroll
    for (int i = 0; i < 4; ++i) {     // 16 rows x 64 cols slots (36 used)
      int e = tid + i * 256;
      int r = e >> 6, col = e & 63;
      if (col < 36) {
        int l = l0 + col - 2;
        float v = 0.0f;
        if ((unsigned)l < (unsigned)LEN) v = srcs[r * LEN + l];
        sBT[col][r] = v;              // transposed store
      }
    }
    __syncthreads();

#pragma unroll
    for (int kk = 0; kk < KW; ++kk) {
#pragma unroll
      for (int r4 = 0; r4 < 4; ++r4) {            // ic quad within chunk
        const int kbase = kk * CIN + icc + r4 * 4;
        v2f a = *(const v2f*)(wrow + kbase + 2 * hi);          // 8B pair
        v2f b0 = *(const v2f*)(lds + (n + kk) * SROW + r4 * 4 + 2 * hi);
        v2f b1 = *(const v2f*)(lds + (n + 16 + kk) * SROW + r4 * 4 + 2 * hi);
        acc0 = __builtin_amdgcn_wmma_f32_16x16x4_f32(false, a, false, b0,
                                                     (short)0, acc0, false, false);
        acc1 = __builtin_amdgcn_wmma_f32_16x16x4_f32(false, a, false, b1,
                                                     (short)0, acc1, false, false);
      }
    }
  }

  // epilogue: C layout -> VGPR j holds row j (lanes 0-15) / row j+8 (16-31)
#pragma unroll
  for (int j = 0; j < 8; ++j) {
    int ch = m0 + j + hi * 8;
    float bv = bias[ch];
    float* gr = gates + ((size_t)b * C4 + ch) * LEN + l0 + n;
    gr[0]  = acc0[j] + bv;
    gr[16] = acc1[j] + bv;
  }
}

// ------------------------------------------------------------ GroupNorm stats
// One block per (b, group); slab is contiguous 8192 floats.
__global__ __launch_bounds__(256) void gn_stats(const float* __restrict__ gates,
                                                float* __restrict__ stats) {
  __shared__ float rs[256], rq[256];
  const int blk = blockIdx.x;                 // b*16 + g
  const float* base = gates + (size_t)blk * GSLAB;
  float s = 0.0f, q = 0.0f;
  for (int i = threadIdx.x; i < GSLAB; i += 256) {
    float v = base[i];
    s += v;
    q += v * v;
  }
  rs[threadIdx.x] = s;
  rq[threadIdx.x] = q;
  __syncthreads();
  for (int o = 128; o > 0; o >>= 1) {
    if (threadIdx.x < o) {
      rs[threadIdx.x] += rs[threadIdx.x + o];
      rq[threadIdx.x] += rq[threadIdx.x + o];
    }
    __syncthreads();
  }
  if (threadIdx.x == 0) {
    float mean = rs[0] * (1.0f / GSLAB);
    float var  = rq[0] * (1.0f / GSLAB) - mean * mean;
    stats[2 * blk]     = mean;
    stats[2 * blk + 1] = rsqrtf(fmaxf(var, 0.0f) + 1e-5f);
  }
}

// ---------------------------------------------------- fused GN + LSTM update
__global__ __launch_bounds__(256) void lstm_point(
    const float* __restrict__ gates, const float* __restrict__ stats,
    const float* __restrict__ gnw, const float* __restrict__ gnb,
    float* __restrict__ h, float* __restrict__ c,
    float* __restrict__ out, float* __restrict__ outHy,
    float* __restrict__ outCy, int t) {
  int idx = blockIdx.x * 256 + threadIdx.x;   // (b, oc, l) flattened
  if (idx >= BSZ * OC * LEN) return;
  int l  = idx & (LEN - 1);
  int oc = (idx >> 8) & (OC - 1);
  int b  = idx >> 15;

  auto norm = [&](int ch) -> float {
    float raw  = gates[((size_t)b * C4 + ch) * LEN + l];
    int g      = ch >> 5;
    float mean = stats[(b * NGRP + g) * 2];
    float rstd = stats[(b * NGRP + g) * 2 + 1];
    return (raw - mean) * rstd * gnw[ch] + gnb[ch];
  };
  float ni = norm(oc);
  float nf = norm(oc + 128);
  float ng = norm(oc + 256);
  float no = norm(oc + 384);

  float si = 1.0f / (1.0f + expf(-ni));
  float sf = 1.0f / (1.0f + expf(-nf));
  float so = 1.0f / (1.0f + expf(-no));

  float cy = sf * c[idx] + si * tanhf(ng);
  float hy = so * tanhf(cy);
  c[idx] = cy;
  h[idx] = hy;
  out[(size_t)t * (BSZ * OC * LEN) + idx] = hy;
  if (t == SEQ - 1) {
    outHy[idx] = hy;
    outCy[idx] = cy;
  }
}

// --------------------------------------------------------------------- host
extern "C" void kernel_launch(void* const* d_in, const int* in_sizes, int n_in,
                              void* d_out, int out_size, void* d_ws,
                              size_t ws_size, hipStream_t stream) {
  const float* x   = (const float*)d_in[0];  // (32,32,64,256)
  const float* cw  = (const float*)d_in[1];  // (512,192,5)
  const float* cb  = (const float*)d_in[2];  // (512)
  const float* gnw = (const float*)d_in[3];  // (512)
  const float* gnb = (const float*)d_in[4];  // (512)

  float* out = (float*)d_out;
  float* ws  = (float*)d_ws;

  float* packW = ws + OFF_PACKW;
  float* h     = ws + OFF_H;
  float* c     = ws + OFF_C;
  float* gates = ws + OFF_GATES;
  float* stats = ws + OFF_STATS;

  float* outHy = out + (size_t)SEQ * BSZ * OC * LEN;
  float* outCy = outHy + (size_t)BSZ * OC * LEN;

  pack_weights<<<(C4 * KTOT + 255) / 256, 256, 0, stream>>>(cw, packW);
  zero_buf<<<(2 * SZ_H + 255) / 256, 256, 0, stream>>>(h, 2 * SZ_H);  // h & c contiguous

  for (int t = 0; t < SEQ; ++t) {
    conv_gates_wmma<<<dim3(C4 / 128, (BSZ * LEN) / 32), 256, 0, stream>>>(
        x, h, packW, cb, gates, t);
    gn_stats<<<BSZ * NGRP, 256, 0, stream>>>(gates, stats);
    lstm_point<<<(BSZ * OC * LEN) / 256, 256, 0, stream>>>(
        gates, stats, gnw, gnb, h, c, out, outHy, outCy, t);
  }
}